// PublicNodeEncoder_11596411699547
// MI455X (gfx1250) — compile-verified
//
#include <hip/hip_runtime.h>

typedef __attribute__((ext_vector_type(2))) float v2f;
typedef __attribute__((ext_vector_type(8))) float v8f;

#define WAVES_PER_BLOCK 4
#define MBLK 4   // 16-row M tiles per wave (64-row strip): B frags amortized over 4 WMMAs

// ---------------- degree / normalization ----------------
__global__ void degree_kernel(const int* __restrict__ dst, float* __restrict__ deg, int E) {
    int e = blockIdx.x * blockDim.x + threadIdx.x;
    if (e < E) atomicAdd(&deg[dst[e]], 1.0f);
}

// in-place: deg -> dinv = rsqrt(deg + 1)   (+1 = self loop)
__global__ void dinv_kernel(float* __restrict__ deg, int n) {
    int i = blockIdx.x * blockDim.x + threadIdx.x;
    if (i < n) deg[i] = rsqrtf(deg[i] + 1.0f);
}

// ---------------- fp32 WMMA GEMM: C[M,N] = A[M,K] * B[K,N] (+bias) ----------------
// One wave computes a 64x16 strip (4 stacked 16x16 tiles) via V_WMMA_F32_16X16X4_F32.
__global__ void gemm_wmma_f32(const float* __restrict__ A, const float* __restrict__ B,
                              const float* __restrict__ bias, float* __restrict__ C,
                              int M, int N, int K) {
    const int lane   = threadIdx.x & 31;
    const int wave   = threadIdx.x >> 5;
    const int tileM0 = (blockIdx.x * WAVES_PER_BLOCK + wave) * (MBLK * 16);
    const int tileN  = blockIdx.y << 4;
    if (tileM0 >= M) return;  // whole-wave exit: EXEC stays all-1s for every WMMA

    const int rlo   = lane & 15;
    const int khalf = (lane >> 4) << 1;             // lanes 0-15: K+0/1, lanes 16-31: K+2/3
    const int col   = tileN + (lane & 15);
    const int bcol  = (col < N) ? col : (N - 1);    // clamped; garbage cols never stored

    const float* Ap[MBLK];
#pragma unroll
    for (int t = 0; t < MBLK; ++t) {
        int r = tileM0 + t * 16 + rlo;
        if (r >= M) r = M - 1;                      // clamped; garbage rows never stored
        Ap[t] = A + (size_t)r * K;
    }

    v8f acc[MBLK] = {};
    for (int k0 = 0; k0 < K; k0 += 4) {
        const int kk = k0 + khalf;
        v2f b;                                      // B frag: VGPR j = rows K=j / K=j+2
        b[0] = B[kk * N + bcol];
        b[1] = B[(kk + 1) * N + bcol];
#pragma unroll
        for (int t = 0; t < MBLK; ++t) {
            v2f a;                                  // A frag: lane=row, VGPRs = K pair
            a[0] = Ap[t][kk];
            a[1] = Ap[t][kk + 1];
            acc[t] = __builtin_amdgcn_wmma_f32_16x16x4_f32(
                /*neg_a=*/false, a, /*neg_b=*/false, b,
                /*c_mod=*/(short)0, acc[t], /*reuse_a=*/false, /*reuse_b=*/false);
        }
    }

    // C/D layout: VGPR j -> row base+j (lanes 0-15) and base+j+8 (lanes 16-31)
    if (col < N) {
        const float bv = bias ? bias[col] : 0.0f;
#pragma unroll
        for (int t = 0; t < MBLK; ++t) {
            const int crow = tileM0 + t * 16 + ((lane >> 4) << 3);
#pragma unroll
            for (int j = 0; j < 8; ++j) {
                const int r = crow + j;
                if (r < M) C[(size_t)r * N + col] = acc[t][j] + bv;
            }
        }
    }
}

// ---------------- edge scatter-aggregate (64 features, float4 per lane) ----------------
// agg[dst] += H[src] * dinv[src]*dinv[dst] ; atomics resolve in 192MB L2 (buffer = 25.6MB)
__global__ void edge_agg_kernel(const int* __restrict__ src, const int* __restrict__ dst,
                                const float* __restrict__ dinv, const float* __restrict__ H,
                                float* __restrict__ agg, int E) {
    int t = blockIdx.x * blockDim.x + threadIdx.x;
    int e = t >> 4;        // 16 lanes per edge
    int q = t & 15;        // float4 chunk within 64 features
    if (e >= E) return;
    int s = src[e], d = dst[e];
    float w = dinv[s] * dinv[d];
    const float4 v = *(const float4*)(H + (s << 6) + (q << 2));
    float* ap = agg + (d << 6) + (q << 2);
    atomicAdd(ap + 0, v.x * w);
    atomicAdd(ap + 1, v.y * w);
    atomicAdd(ap + 2, v.z * w);
    atomicAdd(ap + 3, v.w * w);
}

// ---------------- self-loop + bias + ReLU (in place on agg) ----------------
__global__ void finalize_kernel(float* __restrict__ agg, const float* __restrict__ H,
                                const float* __restrict__ dinv, const float* __restrict__ bias,
                                int total) {
    int idx = blockIdx.x * blockDim.x + threadIdx.x;
    if (idx >= total) return;
    int node = idx >> 6;
    int f    = idx & 63;
    float w  = dinv[node];
    float v  = agg[idx] + H[idx] * w * w + bias[f];
    agg[idx] = v > 0.0f ? v : 0.0f;
}

// ---------------- driver ----------------
extern "C" void kernel_launch(void* const* d_in, const int* in_sizes, int n_in,
                              void* d_out, int out_size, void* d_ws, size_t ws_size,
                              hipStream_t stream) {
    const float* x  = (const float*)d_in[0];
    const int*   ei = (const int*)d_in[1];
    const float* W1 = (const float*)d_in[2];
    const float* b1 = (const float*)d_in[3];
    const float* W2 = (const float*)d_in[4];
    const float* b2 = (const float*)d_in[5];
    const float* Wh = (const float*)d_in[6];
    const float* bh = (const float*)d_in[7];

    const int NN = in_sizes[0] / 128;   // 100000 nodes
    const int E  = in_sizes[1] / 2;     // 1.6M edges
    const int* src = ei;
    const int* dst = ei + E;

    // Workspace layout (A2 aliases H1: H1 dies after layer-1 finalize, A2 is
    // first touched — zeroed — strictly after that point in stream order).
    float* ws = (float*)d_ws;
    size_t o = 0;
    float* dinv = ws + o; o += ((size_t)NN + 255) & ~(size_t)255;
    float* H1   = ws + o; o += (size_t)NN * 64;
    float* A1   = ws + o; o += (size_t)NN * 64;
    float* H2   = ws + o; o += (size_t)NN * 64;
    float* A2   = H1;                      // alias — total footprint ~77MB
    (void)ws_size; (void)n_in; (void)out_size;

    hipMemsetAsync(dinv, 0, (size_t)NN * sizeof(float), stream);
    hipMemsetAsync(A1,   0, (size_t)NN * 64 * sizeof(float), stream);

    // degrees -> dinv
    degree_kernel<<<(E + 255) / 256, 256, 0, stream>>>(dst, dinv, E);
    dinv_kernel<<<(NN + 255) / 256, 256, 0, stream>>>(dinv, NN);

    const int MT     = (NN + 15) / 16;                       // 16-row tiles
    const int STRIPS = (MT + MBLK - 1) / MBLK;               // 64-row strips
    const dim3 blk(32 * WAVES_PER_BLOCK);
    const dim3 gN64((STRIPS + WAVES_PER_BLOCK - 1) / WAVES_PER_BLOCK, 64 / 16);
    const dim3 gN40((STRIPS + WAVES_PER_BLOCK - 1) / WAVES_PER_BLOCK, (40 + 15) / 16);
    const int eth = E * 16;
    const int ft  = NN * 64;

    // layer 1
    gemm_wmma_f32<<<gN64, blk, 0, stream>>>(x, W1, nullptr, H1, NN, 64, 128);
    edge_agg_kernel<<<(eth + 255) / 256, 256, 0, stream>>>(src, dst, dinv, H1, A1, E);
    finalize_kernel<<<(ft + 255) / 256, 256, 0, stream>>>(A1, H1, dinv, b1, ft);

    // layer 2 (A2 buffer = old H1 space, zeroed only now that H1 is dead)
    hipMemsetAsync(A2, 0, (size_t)NN * 64 * sizeof(float), stream);
    gemm_wmma_f32<<<gN64, blk, 0, stream>>>(A1, W2, nullptr, H2, NN, 64, 64);
    edge_agg_kernel<<<(eth + 255) / 256, 256, 0, stream>>>(src, dst, dinv, H2, A2, E);
    finalize_kernel<<<(ft + 255) / 256, 256, 0, stream>>>(A2, H2, dinv, b2, ft);

    // head: [NN,64] @ [64,40] + bh -> d_out
    gemm_wmma_f32<<<gN40, blk, 0, stream>>>(A2, Wh, bh, (float*)d_out, NN, 40, 64);
}